// VoxelCollisionCost_19533511262458
// MI455X (gfx1250) — compile-verified
//
#include <hip/hip_runtime.h>

// ---------------------------------------------------------------------------
// VoxelCollisionCost on gfx1250 (MI455X, wave32).
//
//   * one wave32 per (b,h)  -> 32768 waves (8 per 256-thread block)
//   * 4x V_WMMA_F32_16X16X4_F32 homogeneous transforms per wave:
//       A (16x4): rows m = l2*8+i (i<3), A[m][0..2]=rot row, A[m][3]=pos
//       B (4x16): cols n = l2*8+s,      B[0..2][n]=center,  B[3][n]=1.0
//     zero rows in A enforce the 2-link block-diagonal per tile.
//   * D layout: lanes 0-7  -> link 2p   (sphere=lane,   xyz in D[0..2])
//               lanes 24-31-> link 2p+1 (sphere=lane-24, xyz in D[0..2])
//   * per-lane SDF gather (64 MB grid is L2-resident on 192 MB L2)
//   * voxel index uses truncating v_cvt_i32_f32 directly: floor-then-clamp
//     and trunc-then-clamp agree for all inputs (negatives clamp to 0).
//   * per-link max over 8 spheres: 3x single-instruction ds_swizzle SWAPX
//   * final link-sum combine via v_readlane.
// ---------------------------------------------------------------------------

typedef __attribute__((ext_vector_type(2))) float v2f;
typedef __attribute__((ext_vector_type(8))) float v8f;

#define N_LINKS   8
#define N_SPH     8
#define GRID_N    256

// ds_swizzle group-of-32 imm: offset[14:10]=xor, [9:5]=or, [4:0]=and
#define SWZ_X1 0x041F
#define SWZ_X2 0x081F
#define SWZ_X4 0x101F

#define SWZ_MAX(v, imm)                                                        \
    fmaxf((v), __int_as_float(                                                 \
                   __builtin_amdgcn_ds_swizzle(__float_as_int(v), (imm))))

__global__ __launch_bounds__(256)
void voxcost_wmma_kernel(const float* __restrict__ posg,   // [BH,8,3]
                         const float* __restrict__ rotg,   // [BH,8,3,3]
                         const float* __restrict__ ctrg,   // [8,8,3]
                         const float* __restrict__ radg,   // [8,8]
                         const float* __restrict__ sdf,    // [256^3]
                         const float* __restrict__ wgt,    // [1]
                         float* __restrict__ out,          // [BH]
                         int BH)
{
    __shared__ float s_ctr[N_LINKS * N_SPH * 3];   // 768 B
    __shared__ float s_rad[N_LINKS * N_SPH];       // 256 B

    const int tid = threadIdx.x;
    if (tid < N_LINKS * N_SPH * 3) s_ctr[tid] = ctrg[tid];
    if (tid < N_LINKS * N_SPH)     s_rad[tid] = radg[tid];
    __syncthreads();

    const int wave = blockIdx.x * (blockDim.x >> 5) + (tid >> 5);
    const int lane = tid & 31;
    if (wave >= BH) return;                 // wave-uniform

    const float* rotB = rotg + (size_t)wave * (N_LINKS * 9);
    const float* posB = posg + (size_t)wave * (N_LINKS * 3);

    // A-matrix lane decode (16x4 f32: lanes<16 hold K={0,1}, lanes>=16 K={2,3})
    const int m      = lane & 15;           // A row
    const int khalf  = lane >> 4;           // which K-pair this lane supplies
    const int l2     = m >> 3;              // link-in-pair from row
    const int irow   = m & 7;               // coord i (valid if < 3)
    const bool rowOk = (irow < 3);
    const int ri     = rowOk ? irow : 2;    // clamped -> loads always in-bounds
    // B-matrix lane decode
    const int lcol   = (lane & 15) >> 3;    // link-in-pair from col
    const int scol   = lane & 7;            // sphere index

    float acc = 0.0f;   // lanes 0-7: links 0,2,4,6 ; lanes 24-31: links 1,3,5,7

    #pragma unroll
    for (int p = 0; p < 4; ++p) {
        const int lrow = 2 * p + l2;        // link for this lane's A row
        const int lc   = 2 * p + lcol;      // link for this lane's B col

        // ---- unconditional loads (no EXEC-masked branches), then cndmask
        const float* rr = rotB + lrow * 9 + ri * 3;
        const float r0 = rr[0];
        const float r1 = rr[1];
        const float r2 = rr[2];
        const float pp = posB[lrow * 3 + ri];

        v2f Amat;
        {
            const float a0 = (khalf == 0) ? r0 : r2;
            const float a1 = (khalf == 0) ? r1 : pp;
            Amat[0] = rowOk ? a0 : 0.0f;
            Amat[1] = rowOk ? a1 : 0.0f;
        }
        v2f Bmat;
        {
            const float* cc = s_ctr + (lc * N_SPH + scol) * 3;
            const float c0 = cc[0], c1 = cc[1], c2 = cc[2];
            Bmat[0] = (khalf == 0) ? c0 : c2;
            Bmat[1] = (khalf == 0) ? c1 : 1.0f;
        }

        v8f Cz = {};
        v8f D = __builtin_amdgcn_wmma_f32_16x16x4_f32(
                    /*neg_a=*/false, Amat, /*neg_b=*/false, Bmat,
                    /*c_mod=*/(short)0, Cz, /*reuse_a=*/false, /*reuse_b=*/false);

        const float px = D[0];
        const float py = D[1];
        const float pz = D[2];

        // voxel index: trunc-cast + med3 clamp == floor + clip here
        // (negative values: floor<=trunc<=0, both clamp to 0)
        int ix = (int)((px + 1.28f) * 100.0f);
        int iy = (int)((py + 1.28f) * 100.0f);
        int iz = (int)((pz + 1.28f) * 100.0f);
        ix = min(max(ix, 0), GRID_N - 1);
        iy = min(max(iy, 0), GRID_N - 1);
        iz = min(max(iz, 0), GRID_N - 1);

        const float g = sdf[(ix << 16) | (iy << 8) | iz];   // L2-resident gather

        // penetration for this lane's sphere (lanes 8-23 garbage; never read)
        float pen = s_rad[(2 * p + khalf) * N_SPH + scol] - g;

        // max over the 8 spheres of the link: 3x ds_swizzle SWAPX butterflies
        pen = SWZ_MAX(pen, SWZ_X1);
        pen = SWZ_MAX(pen, SWZ_X2);
        pen = SWZ_MAX(pen, SWZ_X4);

        // clip(max + (-0.01), 0, 0.5) / 0.25 ; every lane in the 8-group now
        // holds the group max, so accumulate unconditionally.
        float r = pen - 0.01f;
        r = fminf(fmaxf(r, 0.0f), 0.5f) * 4.0f;
        acc += r;
    }

    // lane 0 carries sum over links {0,2,4,6}, lane 24 over {1,3,5,7}
    const float tot =
        __int_as_float(__builtin_amdgcn_readlane(__float_as_int(acc), 0)) +
        __int_as_float(__builtin_amdgcn_readlane(__float_as_int(acc), 24));
    if (lane == 0) out[wave] = wgt[0] * tot;
}

extern "C" void kernel_launch(void* const* d_in, const int* in_sizes, int n_in,
                              void* d_out, int out_size, void* d_ws, size_t ws_size,
                              hipStream_t stream)
{
    const float* posg = (const float*)d_in[0];   // link_pos_seq  [B,H,8,3]
    const float* rotg = (const float*)d_in[1];   // link_rot_seq  [B,H,8,3,3]
    const float* ctrg = (const float*)d_in[2];   // sphere_centers [8,8,3]
    const float* radg = (const float*)d_in[3];   // sphere_radii   [8,8]
    const float* sdfg = (const float*)d_in[4];   // sdf_grid       [256^3]
    const float* wgt  = (const float*)d_in[5];   // weight         [1]
    float* out = (float*)d_out;                  // [B,H]

    const int BH = in_sizes[0] / (N_LINKS * 3);  // B*H
    const int wavesPerBlock = 8;                 // 256 threads = 8 wave32
    const int blocks = (BH + wavesPerBlock - 1) / wavesPerBlock;

    voxcost_wmma_kernel<<<blocks, 256, 0, stream>>>(posg, rotg, ctrg, radg,
                                                    sdfg, wgt, out, BH);
}